// GptOssSparseMoeBlock_30236569763903
// MI455X (gfx1250) — compile-verified
//
#include <hip/hip_runtime.h>
#include <hip/hip_bf16.h>
#include <math.h>

typedef float v2f __attribute__((ext_vector_type(2)));
typedef float v8f __attribute__((ext_vector_type(8)));

#define T_TOK   1024
#define H_DIM   1024
#define M_DIM   4096
#define E_NUM   16
#define TWO_M   8192
#define ALPHA_C 1.702f
#define LIMIT_C 7.0f

#define TILE_ROWS 32
#define MAX_ROWS  (2*T_TOK + E_NUM*TILE_ROWS)   // 2560
#define MAX_TILES (MAX_ROWS / TILE_ROWS)        // 80
#define KC   64
#define XLD  (KC + 4)                            // LDS pad: stride 68 floats (16B aligned rows)

// workspace layout (bytes)
#define WS_SCALARS_BYTES 256                     // ints: [0..15]=counts, [16..32]=offsets, [33]=total_rows
#define STAGED_TOK_OFF   (WS_SCALARS_BYTES)                        // int[E*T]
#define STAGED_W_OFF     (STAGED_TOK_OFF + E_NUM*T_TOK*4)          // float[E*T]
#define ROW_TOK_OFF      (STAGED_W_OFF + E_NUM*T_TOK*4)            // int[MAX_ROWS]
#define ROW_W_OFF        (ROW_TOK_OFF + MAX_ROWS*4)                // float[MAX_ROWS]
#define ROW_E_OFF        (ROW_W_OFF + MAX_ROWS*4)                  // int[MAX_ROWS]
#define GATED_OFF        162048                                    // float[MAX_ROWS*M_DIM], 256B aligned

// ---- CDNA5 async global->LDS copy (ASYNCcnt-tracked), with safe fallback ----
#if defined(__has_builtin)
#if __has_builtin(__builtin_amdgcn_global_load_async_to_lds_b128) && \
    __has_builtin(__builtin_amdgcn_s_wait_asynccnt)
#define HAVE_ASYNC_LDS 1
#endif
#endif

typedef int async_v4i __attribute__((vector_size(16)));
typedef __attribute__((address_space(1))) async_v4i* g_v4i_ptr;
typedef __attribute__((address_space(3))) async_v4i* l_v4i_ptr;

__device__ __forceinline__ void async_copy16(float* lds_dst, const float* gsrc) {
#ifdef HAVE_ASYNC_LDS
    __builtin_amdgcn_global_load_async_to_lds_b128(
        (g_v4i_ptr)(void*)const_cast<float*>(gsrc),
        (l_v4i_ptr)(void*)lds_dst,
        0, 0);
#else
    *(float4*)lds_dst = *(const float4*)gsrc;
#endif
}

__device__ __forceinline__ void wait_async_lds() {
#ifdef HAVE_ASYNC_LDS
    __builtin_amdgcn_s_wait_asynccnt(0);
#endif
}

// ---------------- Router: logits, top-2 softmax, expert lists ----------------
__global__ __launch_bounds__(128) void router_kernel(
    const float* __restrict__ x, const float* __restrict__ rw,
    const float* __restrict__ rb, float* __restrict__ scores,
    int* __restrict__ counts, int* __restrict__ staged_tok,
    float* __restrict__ staged_w)
{
    int t = blockIdx.x;
    int tid = threadIdx.x;
    int e = tid & 15, j = tid >> 4;          // 16 experts x 8 chunks
    const float* xr = x + (size_t)t * H_DIM;
    float p = 0.f;
    int h0 = j * 128;
    for (int h = h0; h < h0 + 128; ++h)
        p += xr[h] * rw[h * E_NUM + e];

    __shared__ float red[8][16];
    __shared__ float lg[16];
    red[j][e] = p;
    __syncthreads();
    if (tid < 16) {
        float s = rb[tid];
        #pragma unroll
        for (int k = 0; k < 8; ++k) s += red[k][tid];
        lg[tid] = s;
    }
    __syncthreads();
    if (tid == 0) {
        int i1 = 0; float v1 = lg[0];
        for (int k = 1; k < 16; ++k) if (lg[k] > v1) { v1 = lg[k]; i1 = k; }
        int i2 = (i1 == 0) ? 1 : 0; float v2 = -3.4e38f;
        for (int k = 0; k < 16; ++k) if (k != i1 && lg[k] > v2) { v2 = lg[k]; i2 = k; }
        float w1 = 1.f / (1.f + __expf(v2 - v1));
        float w2 = 1.f - w1;
        scores[t * E_NUM + i1] = w1;
        scores[t * E_NUM + i2] = w2;
        int p1 = atomicAdd(&counts[i1], 1);
        staged_tok[i1 * T_TOK + p1] = t; staged_w[i1 * T_TOK + p1] = w1;
        int p2 = atomicAdd(&counts[i2], 1);
        staged_tok[i2 * T_TOK + p2] = t; staged_w[i2 * T_TOK + p2] = w2;
    }
}

// ---------------- Compaction: padded per-expert row ranges ----------------
__global__ __launch_bounds__(256) void compact_kernel(
    int* __restrict__ ws0, const int* __restrict__ staged_tok,
    const float* __restrict__ staged_w, int* __restrict__ row_tok,
    float* __restrict__ row_w, int* __restrict__ row_e)
{
    __shared__ int s_off[E_NUM + 1];
    __shared__ int s_cnt[E_NUM];
    if (threadIdx.x == 0) {
        int off = 0;
        for (int e = 0; e < E_NUM; ++e) {
            s_off[e] = off;
            int c = ws0[e];
            s_cnt[e] = c;
            off += (c + TILE_ROWS - 1) & ~(TILE_ROWS - 1);
        }
        s_off[E_NUM] = off;
        ws0[33] = off;
        for (int e = 0; e <= E_NUM; ++e) ws0[16 + e] = s_off[e];
    }
    __syncthreads();
    for (int e = 0; e < E_NUM; ++e) {
        int base = s_off[e];
        int n = s_off[e + 1] - base;
        int c = s_cnt[e];
        for (int i = threadIdx.x; i < n; i += 256) {
            int r = base + i;
            if (i < c) { row_tok[r] = staged_tok[e * T_TOK + i]; row_w[r] = staged_w[e * T_TOK + i]; }
            else       { row_tok[r] = 0;                          row_w[r] = 0.f; }
            row_e[r] = e;
        }
    }
}

// ---------------- Pass A: gate_up GEMM + GLU activation ----------------
// block: 256 thr = 8 waves; 32 rows x 128 gate_up cols; K = H_DIM
__global__ __launch_bounds__(256) void gateup_kernel(
    const float* __restrict__ x, const float* __restrict__ gup,
    const float* __restrict__ gup_b, const int* __restrict__ ws0,
    const int* __restrict__ row_tok, const int* __restrict__ row_e,
    float* __restrict__ gated)
{
    int row0 = blockIdx.x * TILE_ROWS;
    if (row0 >= ws0[33]) return;
    int e = row_e[row0];
    int tid = threadIdx.x;
    int lane = tid & 31;
    int wv = tid >> 5;
    int c0 = blockIdx.y * 128 + wv * 16;     // gate_up column base for this wave

    __shared__ float sx[2][TILE_ROWS][XLD];  // double-buffered A tile
    __shared__ float sep[8][TILE_ROWS][16];
    __shared__ int   stok[TILE_ROWS];

    if (tid < TILE_ROWS) stok[tid] = row_tok[row0 + tid];
    __syncthreads();

    // static staging assignment: thread -> rows (sr, sr+16), cols [sc, sc+4)
    int sr = tid >> 4;              // 0..15
    int sc = (tid & 15) * 4;        // 0..60
    const float* ga = x + (size_t)stok[sr] * H_DIM + sc;        // token bases hoisted once
    const float* gb = x + (size_t)stok[sr + 16] * H_DIM + sc;

    const float* W = gup + (size_t)e * H_DIM * TWO_M;
    v8f acc0 = {}; v8f acc1 = {};
    int n = lane & 15;
    int khalf = (lane >> 4) << 1;            // 0 or 2

    // stage chunk 0
    async_copy16(&sx[0][sr][sc],      ga);
    async_copy16(&sx[0][sr + 16][sc], gb);
    wait_async_lds();
    __syncthreads();

    const int NCH = H_DIM / KC;
    for (int ch = 0; ch < NCH; ++ch) {
        int k0 = ch * KC;
        int cur = ch & 1, nxt = cur ^ 1;
        if (ch + 1 < NCH) {
            async_copy16(&sx[nxt][sr][sc],      ga + k0 + KC);
            async_copy16(&sx[nxt][sr + 16][sc], gb + k0 + KC);
        }
        #pragma unroll 4
        for (int kk = 0; kk < KC; kk += 4) {
            v2f a0 = *(const v2f*)&sx[cur][n][kk + khalf];
            v2f a1 = *(const v2f*)&sx[cur][n + 16][kk + khalf];
            const float* wp = W + (size_t)(k0 + kk + khalf) * TWO_M + c0 + n;
            v2f b; b.x = wp[0]; b.y = wp[TWO_M];
            acc0 = __builtin_amdgcn_wmma_f32_16x16x4_f32(false, a0, false, b, (short)0, acc0, false, false);
            acc1 = __builtin_amdgcn_wmma_f32_16x16x4_f32(false, a1, false, b, (short)0, acc1, false, false);
        }
        wait_async_lds();
        __syncthreads();
    }

    // epilogue: stash accumulators to LDS, deinterleave gate/up, activate
    int rhalf = (lane >> 4) << 3;            // 0 or 8
    #pragma unroll
    for (int v = 0; v < 8; ++v) {
        sep[wv][v + rhalf][n]      = acc0[v];
        sep[wv][v + rhalf + 16][n] = acc1[v];
    }
    __syncthreads();
    {
        int r = lane;                         // one row per lane
        int m0 = c0 >> 1;                     // 8 gated cols per wave
        float* gp = gated + (size_t)(row0 + r) * M_DIM + m0;
        #pragma unroll
        for (int jj = 0; jj < 8; ++jj) {
            float g = sep[wv][r][2 * jj]     + gup_b[(size_t)e * TWO_M + c0 + 2 * jj];
            float u = sep[wv][r][2 * jj + 1] + gup_b[(size_t)e * TWO_M + c0 + 2 * jj + 1];
            g = fminf(g, LIMIT_C);
            g = fmaxf(g, -1e9f);
            u = fminf(fmaxf(u, -LIMIT_C), LIMIT_C);
            float glu = g / (1.f + __expf(-ALPHA_C * g));
            gp[jj] = (u + 1.f) * glu;
        }
    }
}

// ---------------- Pass B: down GEMM + weighted scatter-add ----------------
// block: 256 thr = 8 waves; 32 rows x 128 H cols; K = M_DIM
__global__ __launch_bounds__(256) void down_kernel(
    const float* __restrict__ gated, const float* __restrict__ dw,
    const float* __restrict__ db, const int* __restrict__ ws0,
    const int* __restrict__ row_tok, const float* __restrict__ row_w,
    const int* __restrict__ row_e, float* __restrict__ out)
{
    int row0 = blockIdx.x * TILE_ROWS;
    if (row0 >= ws0[33]) return;
    int e = row_e[row0];
    int tid = threadIdx.x, lane = tid & 31, wv = tid >> 5;
    int h0 = blockIdx.y * 128 + wv * 16;

    __shared__ float sg[2][TILE_ROWS][XLD];  // double-buffered A tile
    __shared__ int   stok[TILE_ROWS];
    __shared__ float srw[TILE_ROWS];
    if (tid < TILE_ROWS) { stok[tid] = row_tok[row0 + tid]; srw[tid] = row_w[row0 + tid]; }
    __syncthreads();

    int sr = tid >> 4;
    int sc = (tid & 15) * 4;
    const float* ga = gated + (size_t)(row0 + sr) * M_DIM + sc;
    const float* gb = gated + (size_t)(row0 + sr + 16) * M_DIM + sc;

    const float* W = dw + (size_t)e * M_DIM * H_DIM;
    v8f acc0 = {}; v8f acc1 = {};
    int n = lane & 15;
    int khalf = (lane >> 4) << 1;

    async_copy16(&sg[0][sr][sc],      ga);
    async_copy16(&sg[0][sr + 16][sc], gb);
    wait_async_lds();
    __syncthreads();

    const int NCH = M_DIM / KC;
    for (int ch = 0; ch < NCH; ++ch) {
        int k0 = ch * KC;
        int cur = ch & 1, nxt = cur ^ 1;
        if (ch + 1 < NCH) {
            async_copy16(&sg[nxt][sr][sc],      ga + k0 + KC);
            async_copy16(&sg[nxt][sr + 16][sc], gb + k0 + KC);
        }
        #pragma unroll 4
        for (int kk = 0; kk < KC; kk += 4) {
            v2f a0 = *(const v2f*)&sg[cur][n][kk + khalf];
            v2f a1 = *(const v2f*)&sg[cur][n + 16][kk + khalf];
            const float* wp = W + (size_t)(k0 + kk + khalf) * H_DIM + h0 + n;
            v2f b; b.x = wp[0]; b.y = wp[H_DIM];
            acc0 = __builtin_amdgcn_wmma_f32_16x16x4_f32(false, a0, false, b, (short)0, acc0, false, false);
            acc1 = __builtin_amdgcn_wmma_f32_16x16x4_f32(false, a1, false, b, (short)0, acc1, false, false);
        }
        wait_async_lds();
        __syncthreads();
    }

    float bias = db[(size_t)e * H_DIM + h0 + n];
    int rhalf = (lane >> 4) << 3;
    #pragma unroll
    for (int v = 0; v < 8; ++v) {
        int r0 = v + rhalf;
        unsafeAtomicAdd(out + (size_t)stok[r0] * H_DIM + h0 + n, srw[r0] * (acc0[v] + bias));
        int r1 = r0 + 16;
        unsafeAtomicAdd(out + (size_t)stok[r1] * H_DIM + h0 + n, srw[r1] * (acc1[v] + bias));
    }
}

extern "C" void kernel_launch(void* const* d_in, const int* in_sizes, int n_in,
                              void* d_out, int out_size, void* d_ws, size_t ws_size,
                              hipStream_t stream) {
    (void)in_sizes; (void)n_in; (void)out_size; (void)ws_size;
    const float* x    = (const float*)d_in[0];   // (B,S,H) = (2,512,1024)
    const float* rw   = (const float*)d_in[1];   // (H,E)
    const float* rb   = (const float*)d_in[2];   // (E)
    const float* gup  = (const float*)d_in[3];   // (E,H,2M)
    const float* gupb = (const float*)d_in[4];   // (E,2M)
    const float* dwn  = (const float*)d_in[5];   // (E,M,H)
    const float* dwnb = (const float*)d_in[6];   // (E,H)

    float* out_final  = (float*)d_out;                                // T*H
    float* out_scores = out_final + (size_t)T_TOK * H_DIM;            // T*E

    char* ws = (char*)d_ws;
    int*   ws0        = (int*)ws;
    int*   staged_tok = (int*)(ws + STAGED_TOK_OFF);
    float* staged_w   = (float*)(ws + STAGED_W_OFF);
    int*   row_tok    = (int*)(ws + ROW_TOK_OFF);
    float* row_w      = (float*)(ws + ROW_W_OFF);
    int*   row_e      = (int*)(ws + ROW_E_OFF);
    float* gated      = (float*)(ws + GATED_OFF);

    (void)hipMemsetAsync(d_ws, 0, WS_SCALARS_BYTES, stream);
    (void)hipMemsetAsync(d_out, 0, (size_t)(T_TOK * H_DIM + T_TOK * E_NUM) * sizeof(float), stream);

    router_kernel<<<T_TOK, 128, 0, stream>>>(x, rw, rb, out_scores, ws0, staged_tok, staged_w);
    compact_kernel<<<1, 256, 0, stream>>>(ws0, staged_tok, staged_w, row_tok, row_w, row_e);

    dim3 gA(MAX_TILES, TWO_M / 128);
    gateup_kernel<<<gA, 256, 0, stream>>>(x, gup, gupb, ws0, row_tok, row_e, gated);

    dim3 gB(MAX_TILES, H_DIM / 128);
    down_kernel<<<gB, 256, 0, stream>>>(gated, dwn, dwnb, ws0, row_tok, row_w, row_e, out_final);
}